// CapsuleNetwork_39565238731183
// MI455X (gfx1250) — compile-verified
//
#include <hip/hip_runtime.h>
#include <hip/hip_bf16.h>
#include <math.h>

// ---------------- problem constants ----------------
#define B_SZ   512
#define L_SZ   512
#define E_SZ   128
#define NKF    32      // conv filters
#define DCAP   16
#define CCAP   64
#define NCC    128
#define H2     127
#define H4     125
#define H8     121
#define LTOT   373     // H2+H4+H8
#define NIN    11936   // NKF*LTOT
#define PADW   128     // padded per-width section in nets_pad
#define PSTR   (3*PADW) // per-(b,o) stride of nets_pad = 384
#define EPS_BN 1e-5f

typedef __attribute__((ext_vector_type(2))) float v2f;
typedef __attribute__((ext_vector_type(8))) float v8f;

// ---------------- fp32 WMMA helper (V_WMMA_F32_16X16X4_F32) ----------------
__device__ __forceinline__ v8f wmma_f32_16x16x4(v2f a, v2f b, v8f c) {
#if defined(__gfx1250__) && __has_builtin(__builtin_amdgcn_wmma_f32_16x16x4_f32)
    return __builtin_amdgcn_wmma_f32_16x16x4_f32(false, a, false, b, (short)0, c,
                                                 false, false);
#else
    c[0] += a.x * b.x + a.y * b.y;
    return c;
#endif
}

// =====================================================================
// K1: fused embedding-gather + three conv1d (widths 2/4/8) + ReLU.
// 1 workgroup per b; 512 threads = 16 waves; emb block (256KB) in LDS.
// Width specialized at compile time -> constant trip counts, unrollable.
// =====================================================================
template<int KLOG>
__device__ __forceinline__ void conv_width_tile(const float* __restrict__ embs,
                                                const float* __restrict__ w,
                                                const float* __restrict__ bias,
                                                float* __restrict__ nets_pad,
                                                int b, int widx, int mt, int nt,
                                                int lsub, int khalf) {
    const int Ktot = L_SZ << KLOG;        // compile-time: 1024/2048/4096
    const int kw   = 1 << KLOG;
    const int h0   = mt * 16;             // mt/nt are SGPRs (readfirstlane'd)
    const int n0   = nt * 16;
    const float* wrow = w + (size_t)(n0 + lsub) * Ktot;

    v8f acc = {};
#pragma unroll 8
    for (int kk0 = 0; kk0 < Ktot; kk0 += 4) {
        const int kkA = kk0 + 2 * khalf;           // even -> pair shares c-row
        const int c0  = kkA >> KLOG;
        const int t0  = kkA & (kw - 1);
        v2f a, bb;
        const float* ap = embs + c0 * E_SZ + h0 + lsub + t0;
        a.x  = ap[0];                               // A[m][kk]   = emb[b,c,h+t]
        a.y  = ap[1];
        bb.x = wrow[kkA];                           // B[kk][n]   = w[o,c,t]
        bb.y = wrow[kkA + 1];
        acc = wmma_f32_16x16x4(a, bb, acc);
    }

    // C tile: VGPR r, lane half -> h = h0 + 8*khalf + r (contiguous), o = lsub
    const int   o  = n0 + lsub;
    const float bv = bias[o];
    float4 q0, q1;
    q0.x = fmaxf(acc[0] + bv, 0.f); q0.y = fmaxf(acc[1] + bv, 0.f);
    q0.z = fmaxf(acc[2] + bv, 0.f); q0.w = fmaxf(acc[3] + bv, 0.f);
    q1.x = fmaxf(acc[4] + bv, 0.f); q1.y = fmaxf(acc[5] + bv, 0.f);
    q1.z = fmaxf(acc[6] + bv, 0.f); q1.w = fmaxf(acc[7] + bv, 0.f);
    float* dst = nets_pad + ((size_t)(b * NKF + o)) * PSTR
                          + widx * PADW + h0 + 8 * khalf;
    *(float4*)(dst)     = q0;
    *(float4*)(dst + 4) = q1;
}

__global__ void conv_kernel(const int* __restrict__ tokens,
                            const float* __restrict__ emb_table,
                            const float* __restrict__ w2, const float* __restrict__ b2,
                            const float* __restrict__ w4, const float* __restrict__ b4,
                            const float* __restrict__ w8, const float* __restrict__ b8,
                            float* __restrict__ nets_pad) {
    extern __shared__ float smem[];
    float* embs = smem;                       // [512*128]
    const int b    = blockIdx.x;
    const int wv   = __builtin_amdgcn_readfirstlane(threadIdx.x >> 5);  // SGPR
    const int lane = threadIdx.x & 31;

    // gather: one wave per 32 token rows, 32 lanes x float4 = one 512B row
    for (int rr = 0; rr < 32; ++rr) {
        const int r   = wv * 32 + rr;                  // scalar row index
        const int tok = tokens[b * L_SZ + r];          // scalar load
        float4 vv = ((const float4*)(emb_table + (size_t)tok * E_SZ))[lane];
        ((float4*)embs)[r * (E_SZ / 4) + lane] = vv;
    }
    __syncthreads();

    const int lsub  = lane & 15;
    const int khalf = lane >> 4;
    const int mt    = wv >> 1;     // SGPR tile coords, same for all 3 widths
    const int nt    = wv & 1;

    conv_width_tile<1>(embs, w2, b2, nets_pad, b, 0, mt, nt, lsub, khalf);
    conv_width_tile<2>(embs, w4, b4, nets_pad, b, 1, mt, nt, lsub, khalf);
    conv_width_tile<3>(embs, w8, b8, nets_pad, b, 2, mt, nt, lsub, khalf);
}

// =====================================================================
// K2: primary caps (1x1 conv) + squash + reshape-scramble + w_comp GEMM.
// 1 workgroup per b; 256 threads = 8 waves (one NCC m-tile each).
// Chunk length is compile-time (46 x 256 + 1 x 160) -> no guards, unroll.
// =====================================================================
template<int CLEN>
__device__ __forceinline__ v8f comp_chunk(v8f acc, int nbase, int tid,
                                          const float* __restrict__ nets_s,
                                          const float* __restrict__ pw_s,
                                          const float* __restrict__ scale_s,
                                          const float* __restrict__ pb_s,
                                          float* __restrict__ Pchunk,
                                          const float* __restrict__ w_comp,
                                          int m0, int lsub, int khalf) {
    // rebuild P[n,h] = squash(u) after the (DC,32,LTOT)->(N,16) reshape
    for (int q = tid; q < CLEN * 16; q += 256) {
        const int n   = nbase + (q >> 4);
        const int h   = q & 15;
        const int f   = n * 16 + h;
        const int d   = f / NIN;
        const int rem = f - d * NIN;          // = k*LTOT + p
        const int p   = rem % LTOT;
        const int j   = d * 32 + (rem / LTOT);
        float u = pb_s[j];
        const float* pw = pw_s + j * 32;
#pragma unroll
        for (int c = 0; c < 32; ++c) u += nets_s[c * LTOT + p] * pw[c];
        Pchunk[q] = u * scale_s[rem];
    }
    __syncthreads();

#pragma unroll 8
    for (int kk0 = 0; kk0 < CLEN; kk0 += 4) {
        const int kkA = kk0 + 2 * khalf;
        const size_t na0 = (size_t)(nbase + kkA);
        v2f a, bb;
        a.x  = w_comp[na0 * NCC + m0 + lsub];
        a.y  = w_comp[(na0 + 1) * NCC + m0 + lsub];
        bb.x = Pchunk[kkA * 16 + lsub];
        bb.y = Pchunk[(kkA + 1) * 16 + lsub];
        acc = wmma_f32_16x16x4(a, bb, acc);
    }
    __syncthreads();
    return acc;
}

__global__ void prim_comp_kernel(const float* __restrict__ nets_pad,
                                 const float* __restrict__ prim_w,
                                 const float* __restrict__ prim_b,
                                 const float* __restrict__ w_comp,
                                 float* __restrict__ poses) {
    extern __shared__ float smem[];
    float* nets_s  = smem;                 // 11936 (compact: [c][p], p in [0,373))
    float* pw_s    = nets_s + NIN;         // 16384  (512x32)
    float* scale_s = pw_s + 512 * 32;      // 11936
    float* Pchunk  = scale_s + NIN;        // 4096   (<=256 x 16)
    float* pb_s    = Pchunk + 4096;        // 512

    const int b   = blockIdx.x;
    const int tid = threadIdx.x;

    // compact the padded conv output while staging to LDS
    for (int i = tid; i < NIN; i += 256) {
        const int c = i / LTOT, p = i - c * LTOT;
        const int widx = (p < H2) ? 0 : ((p < H2 + H4) ? 1 : 2);
        const int hh   = p - ((widx == 0) ? 0 : ((widx == 1) ? H2 : (H2 + H4)));
        nets_s[i] = nets_pad[((size_t)(b * NKF + c)) * PSTR + widx * PADW + hh];
    }
    for (int i = tid; i < 512 * 32; i += 256) pw_s[i] = prim_w[i];
    for (int i = tid; i < 512; i += 256)      pb_s[i] = prim_b[i];
    __syncthreads();

    // squash scale over the DC axis: s[k,p] = sum_d u[d*32+k, p]^2
    for (int kp = tid; kp < NIN; kp += 256) {
        const int k = kp / LTOT, p = kp - k * LTOT;
        float s = 0.f;
        for (int d = 0; d < DCAP; ++d) {
            const int j = d * 32 + k;
            float u = pb_s[j];
            const float* pw = pw_s + j * 32;
#pragma unroll
            for (int c = 0; c < 32; ++c) u += nets_s[c * LTOT + p] * pw[c];
            s += u * u;
        }
        scale_s[kp] = sqrtf(s) / (0.5f + s);
    }
    __syncthreads();

    const int lane  = tid & 31;
    const int lsub  = lane & 15;
    const int khalf = lane >> 4;
    const int m0    = __builtin_amdgcn_readfirstlane(tid >> 5) * 16;   // SGPR

    v8f acc = {};
    for (int nbase = 0; nbase < 46 * 256; nbase += 256)
        acc = comp_chunk<256>(acc, nbase, tid, nets_s, pw_s, scale_s, pb_s,
                              Pchunk, w_comp, m0, lsub, khalf);
    acc = comp_chunk<160>(acc, 46 * 256, tid, nets_s, pw_s, scale_s, pb_s,
                          Pchunk, w_comp, m0, lsub, khalf);

    for (int r = 0; r < 8; ++r) {
        const int m = m0 + r + 8 * khalf;
        poses[((size_t)b * NCC + m) * DCAP + lsub] = acc[r];
    }
}

// =====================================================================
// K3: u_hat[b,i,c,o] = sum_h w1[i,c,o,h]*poses[b,i,h]; fused squash(axis o).
// =====================================================================
__global__ void uhat_kernel(const float* __restrict__ poses,
                            const float* __restrict__ w1,
                            float* __restrict__ u_hat,
                            float* __restrict__ u_hat_sq) {
    const int idx = blockIdx.x * 256 + threadIdx.x;     // b*8192 + i*64 + c
    const int b = idx >> 13;
    const int r = idx & 8191;
    const int i = r >> 6;
    const int c = r & 63;

    float pp[DCAP];
    const float* pbase = poses + ((size_t)b * NCC + i) * DCAP;
#pragma unroll
    for (int h = 0; h < DCAP; ++h) pp[h] = pbase[h];

    const float* wb = w1 + (size_t)(i * CCAP + c) * DCAP * DCAP;
    float uo[DCAP];
    float s = 0.f;
#pragma unroll
    for (int o = 0; o < DCAP; ++o) {
        float a = 0.f;
#pragma unroll
        for (int h = 0; h < DCAP; ++h) a += wb[o * DCAP + h] * pp[h];
        uo[o] = a;
        s += a * a;
    }
    const float sc = sqrtf(s) / (0.5f + s);
    const size_t ob = (size_t)idx * DCAP;
#pragma unroll
    for (int o = 0; o < DCAP; ++o) {
        u_hat[ob + o]    = uo[o];
        u_hat_sq[ob + o] = uo[o] * sc;
    }
}

// =====================================================================
// K4: 3 iterations of dynamic routing, 1 workgroup (256 thr) per b.
// =====================================================================
__global__ void routing_kernel(const float* __restrict__ u_hat,
                               const float* __restrict__ u_hat_sq,
                               float* __restrict__ out_poses) {
    extern __shared__ float smem[];
    float* bij  = smem;            // 8192
    float* cmat = bij + 8192;      // 8192
    float* vs   = cmat + 8192;     // 1024
    float* rowt = vs + 1024;       // 128
    float* colt = rowt + 128;      // 64

    const int b   = blockIdx.x;
    const int tid = threadIdx.x;
    const float* uh  = u_hat    + (size_t)b * 8192 * DCAP;
    const float* uhs = u_hat_sq + (size_t)b * 8192 * DCAP;

    for (int i = tid; i < 8192; i += 256) bij[i] = 0.f;
    __syncthreads();

    for (int it = 0; it < 3; ++it) {
        if (tid < 128) {                               // softmax over c (axis 2)
            float mx = -1e30f;
            for (int c = 0; c < 64; ++c) mx = fmaxf(mx, bij[tid * 64 + c]);
            float sum = 0.f;
            for (int c = 0; c < 64; ++c) {
                const float e = expf(bij[tid * 64 + c] - mx);
                cmat[tid * 64 + c] = e;
                sum += e;
            }
            rowt[tid] = sum;
        }
        __syncthreads();
        for (int q = tid; q < 8192; q += 256) cmat[q] /= rowt[q >> 6];
        __syncthreads();
        if (tid < 64) {                                // renormalize over i (axis 1)
            float s = 0.f;
            for (int i = 0; i < 128; ++i) s += cmat[i * 64 + tid];
            colt[tid] = s;
        }
        __syncthreads();
        for (int q = tid; q < 8192; q += 256) cmat[q] /= colt[q & 63];
        __syncthreads();

        for (int q = tid; q < 1024; q += 256) {        // v = sum_i c*u_hat
            const int c = q >> 4, o = q & 15;
            float a = 0.f;
            for (int i = 0; i < 128; ++i)
                a += cmat[i * 64 + c] * uh[(size_t)(i * 64 + c) * DCAP + o];
            vs[q] = a;
        }
        __syncthreads();
        if (tid < 64) {                                // squash(v, axis o)
            float s = 0.f;
            for (int o = 0; o < DCAP; ++o) { const float x = vs[tid * 16 + o]; s += x * x; }
            const float sc = sqrtf(s) / (0.5f + s);
            for (int o = 0; o < DCAP; ++o) vs[tid * 16 + o] *= sc;
        }
        __syncthreads();
        if (it < 2) {                                  // b_ij += 1 - ||uhsq - v||^2
            for (int q = tid; q < 8192; q += 256) {
                const int c = q & 63;
                float a = 0.f;
#pragma unroll
                for (int o = 0; o < DCAP; ++o) {
                    const float d = uhs[(size_t)q * DCAP + o] - vs[c * 16 + o];
                    a += d * d;
                }
                bij[q] += 1.f - a;
            }
            __syncthreads();
        }
    }
    for (int q = tid; q < 1024; q += 256)
        out_poses[(size_t)b * 1024 + q] = vs[q];
}

// =====================================================================
// K5a: per-feature batch mean / population variance (1024 features).
// =====================================================================
__global__ void bnstat_kernel(const float* __restrict__ out_poses,
                              float* __restrict__ mean, float* __restrict__ var) {
    const int f = blockIdx.x * 256 + threadIdx.x;
    if (f >= 1024) return;
    float s = 0.f, s2 = 0.f;
    for (int b = 0; b < B_SZ; ++b) {
        const float x = out_poses[(size_t)b * 1024 + f];
        s += x; s2 += x * x;
    }
    const float m = s * (1.f / B_SZ);
    mean[f] = m;
    var[f]  = s2 * (1.f / B_SZ) - m * m;
}

// =====================================================================
// K5b: batchnorm + linear head -> acts (B,64).
// =====================================================================
__global__ void head_kernel(const float* __restrict__ out_poses,
                            const float* __restrict__ mean, const float* __restrict__ var,
                            const float* __restrict__ gamma, const float* __restrict__ beta,
                            const float* __restrict__ lin_w, const float* __restrict__ lin_b,
                            float* __restrict__ out_acts) {
    const int b = blockIdx.x;
    const int j = threadIdx.x;          // 64 threads
    float a = lin_b[j];
    for (int f = 0; f < 1024; ++f) {
        const float x  = out_poses[(size_t)b * 1024 + f];
        const float xn = (x - mean[f]) * rsqrtf(var[f] + EPS_BN) * gamma[f] + beta[f];
        a += xn * lin_w[j * 1024 + f];
    }
    out_acts[(size_t)b * 64 + j] = a;
}

// =====================================================================
extern "C" void kernel_launch(void* const* d_in, const int* in_sizes, int n_in,
                              void* d_out, int out_size, void* d_ws, size_t ws_size,
                              hipStream_t stream) {
    const int*   tokens    = (const int*)  d_in[0];
    const float* emb_table = (const float*)d_in[1];
    const float* w2  = (const float*)d_in[2];  const float* cb2 = (const float*)d_in[3];
    const float* w4  = (const float*)d_in[4];  const float* cb4 = (const float*)d_in[5];
    const float* w8  = (const float*)d_in[6];  const float* cb8 = (const float*)d_in[7];
    const float* prim_w = (const float*)d_in[8];
    const float* prim_b = (const float*)d_in[9];
    const float* w_comp = (const float*)d_in[10];
    const float* w1     = (const float*)d_in[11];
    const float* bn_g   = (const float*)d_in[12];
    const float* bn_b   = (const float*)d_in[13];
    const float* lin_w  = (const float*)d_in[14];
    const float* lin_b  = (const float*)d_in[15];

    float* out_poses = (float*)d_out;                       // B*1024
    float* out_acts  = out_poses + (size_t)B_SZ * 1024;     // B*64

    // workspace carve-up (floats)
    float* ws       = (float*)d_ws;
    float* nets_pad = ws;                                        // B*32*384    = 6,291,456
    float* poses    = nets_pad + (size_t)B_SZ * NKF * PSTR;      // B*128*16    = 1,048,576
    float* u_hat    = poses    + (size_t)B_SZ * NCC * DCAP;      // B*128*64*16 = 67,108,864
    float* u_hat_sq = u_hat    + (size_t)B_SZ * NCC * CCAP * DCAP;
    float* mean     = u_hat_sq + (size_t)B_SZ * NCC * CCAP * DCAP;
    float* var      = mean + 1024;

    const int lds_conv = L_SZ * E_SZ * 4;                                   // 256 KB
    const int lds_prim = (NIN + 512 * 32 + NIN + 4096 + 512) * 4;           // ~175 KB
    const int lds_rout = (8192 + 8192 + 1024 + 128 + 64) * 4;               // ~69 KB
    hipFuncSetAttribute((const void*)conv_kernel,
                        hipFuncAttributeMaxDynamicSharedMemorySize, lds_conv);
    hipFuncSetAttribute((const void*)prim_comp_kernel,
                        hipFuncAttributeMaxDynamicSharedMemorySize, lds_prim);
    hipFuncSetAttribute((const void*)routing_kernel,
                        hipFuncAttributeMaxDynamicSharedMemorySize, lds_rout);

    conv_kernel<<<B_SZ, 512, lds_conv, stream>>>(tokens, emb_table,
                                                 w2, cb2, w4, cb4, w8, cb8, nets_pad);
    prim_comp_kernel<<<B_SZ, 256, lds_prim, stream>>>(nets_pad, prim_w, prim_b,
                                                      w_comp, poses);
    uhat_kernel<<<(B_SZ * NCC * CCAP) / 256, 256, 0, stream>>>(poses, w1, u_hat, u_hat_sq);
    routing_kernel<<<B_SZ, 256, lds_rout, stream>>>(u_hat, u_hat_sq, out_poses);
    bnstat_kernel<<<4, 256, 0, stream>>>(out_poses, mean, var);
    head_kernel<<<B_SZ, 64, 0, stream>>>(out_poses, mean, var, bn_g, bn_b,
                                         lin_w, lin_b, out_acts);
}